// PreprocessLayerTorch_57913339019801
// MI455X (gfx1250) — compile-verified
//
#include <hip/hip_runtime.h>
#include <hip/hip_bf16.h>
#include <math.h>

// ---------------------------------------------------------------------------
// PreprocessLayerTorch (MIC branch) for MI455X / gfx1250.
//
// Outputs (concatenated flat in d_out, all as float):
//   [0,            4n)              onehot   (n,4)
//   [4n,           4n + n*n)        dist     (n,n)   masked MIC distance
//   [4n + n*n,     4n + n*n + 3nn)  shift    (n,n,3) MIC lattice shifts
//
// Roofline: ~576 MB of stores vs ~1 GFLOP of VALU -> pure store-BW bound
// (~25 us at 23.3 TB/s). Kernel design: float4 (b128) non-temporal stores,
// SoA frac planes for b128 neighbor loads, scalar loads for block-uniform
// data, 288K wave32s of occupancy. WMMA (f32 16x16x4) performs the
// reference's actual GEMM  frac = coord @ inv(cell).
// ---------------------------------------------------------------------------

typedef float f4  __attribute__((ext_vector_type(4)));
typedef float v2f __attribute__((ext_vector_type(2)));
typedef float v8f __attribute__((ext_vector_type(8)));

// ---- 3x3 inverse of cell (adjugate/det), single thread --------------------
__global__ void inv3x3_kernel(const float* __restrict__ cell,
                              float* __restrict__ hinv) {
    float c00 = cell[0], c01 = cell[1], c02 = cell[2];
    float c10 = cell[3], c11 = cell[4], c12 = cell[5];
    float c20 = cell[6], c21 = cell[7], c22 = cell[8];
    float det = c00 * (c11 * c22 - c12 * c21)
              - c01 * (c10 * c22 - c12 * c20)
              + c02 * (c10 * c21 - c11 * c20);
    float rd = 1.0f / det;
    hinv[0] = (c11 * c22 - c12 * c21) * rd;
    hinv[1] = (c02 * c21 - c01 * c22) * rd;
    hinv[2] = (c01 * c12 - c02 * c11) * rd;
    hinv[3] = (c12 * c20 - c10 * c22) * rd;
    hinv[4] = (c00 * c22 - c02 * c20) * rd;
    hinv[5] = (c02 * c10 - c00 * c12) * rd;
    hinv[6] = (c10 * c21 - c11 * c20) * rd;
    hinv[7] = (c01 * c20 - c00 * c21) * rd;
    hinv[8] = (c00 * c11 - c01 * c10) * rd;
}

// ---- frac = wrap(coord @ h_inv) via V_WMMA_F32_16X16X4_F32 ----------------
// One wave32 per 16 atoms. A = coord tile (16x4, K=3 used), B = h_inv padded
// to 4x16 (N=0..2 used), C = 0. ISA layouts (CDNA5 ISA):
//   A 16x4:  lanes 0-15 M=0..15; VGPR0 = K0 (lo half) / K2 (hi half),
//            VGPR1 = K1 / K3.
//   B 4x16:  row striped across lanes: VGPR0 = K0 / K2, VGPR1 = K1 / K3,
//            N = lane&15.
//   C 16x16: VGPR r holds M=r (lanes 0-15) and M=r+8 (lanes 16-31), N=lane&15.
// Results are scattered into SoA planes fx/fy/fz for b128 loads later.
__global__ __launch_bounds__(32) void frac_wmma_kernel(
    const float* __restrict__ coord, const float* __restrict__ hinv,
    float* __restrict__ fx, float* __restrict__ fy, float* __restrict__ fz,
    int n) {
    const int lane   = threadIdx.x;
    const int m      = lane & 15;   // A row / C column index base
    const int hi     = lane >> 4;   // half-wave selector
    const int iBase  = blockIdx.x * 16;

    // A operand: coord[iBase+m][K], K = {0,1} in lo half, {2,3=pad} in hi half
    int row = iBase + m;
    if (row >= n) row = n - 1;      // clamp (n is a multiple of 16 here anyway)
    v2f a;
    if (hi == 0) {
        a.x = coord[row * 3 + 0];
        a.y = coord[row * 3 + 1];
    } else {
        a.x = coord[row * 3 + 2];
        a.y = 0.0f;
    }

    // B operand: hinv[K][N], N = lane&15 (only N<3 nonzero)
    const int ncol = m;
    v2f b;
    if (ncol < 3) {
        if (hi == 0) {
            b.x = hinv[0 * 3 + ncol];
            b.y = hinv[1 * 3 + ncol];
        } else {
            b.x = hinv[2 * 3 + ncol];
            b.y = 0.0f;
        }
    } else {
        b.x = 0.0f;
        b.y = 0.0f;
    }

    v8f c = {};
    // D = A x B + 0  -> D[M][N] = sum_K coord[M][K] * hinv[K][N] = raw frac
    c = __builtin_amdgcn_wmma_f32_16x16x4_f32(
            /*neg_a=*/false, a, /*neg_b=*/false, b,
            /*c_mod=*/(short)0, c, /*reuse_a=*/false, /*reuse_b=*/false);

    if (ncol < 3) {
        float* plane = (ncol == 0) ? fx : ((ncol == 1) ? fy : fz);
#pragma unroll
        for (int r = 0; r < 8; ++r) {
            int mrow = iBase + r + 8 * hi;
            if (mrow < n) {
                float v = c[r];
                plane[mrow] = v - floorf(v);   // wrap into [0,1)
            }
        }
    }
}

// ---- onehot: elems vs {1,6,7,8} -------------------------------------------
__global__ __launch_bounds__(256) void onehot_kernel(
    const int* __restrict__ elems, float* __restrict__ out, int n) {
    int i = blockIdx.x * blockDim.x + threadIdx.x;
    if (i >= n) return;
    int e = elems[i];
    f4 o;
    o.x = (e == 1) ? 1.0f : 0.0f;
    o.y = (e == 6) ? 1.0f : 0.0f;
    o.z = (e == 7) ? 1.0f : 0.0f;
    o.w = (e == 8) ? 1.0f : 0.0f;
    *(f4*)(out + (size_t)i * 4) = o;   // 16B aligned: i*4 floats
}

// ---- main pairwise kernel: store-bandwidth bound --------------------------
// grid = (ceil(n/4/256), n); each thread owns 4 consecutive j for row i.
// All stores are 16B-aligned global_store_b128 with NT temporal hint.
__global__ __launch_bounds__(256) void pair_kernel(
    const float* __restrict__ cell,
    const float* __restrict__ fx, const float* __restrict__ fy,
    const float* __restrict__ fz,
    float* __restrict__ dist_out, float* __restrict__ shift_out,
    int n, float rc2) {
    const int i  = blockIdx.y;
    const int jq = blockIdx.x * blockDim.x + threadIdx.x;
    const int j0 = jq * 4;
    if (j0 >= n) return;

    // Block-uniform scalar loads
    const float c00 = cell[0], c01 = cell[1], c02 = cell[2];
    const float c10 = cell[3], c11 = cell[4], c12 = cell[5];
    const float c20 = cell[6], c21 = cell[7], c22 = cell[8];
    const float fix = fx[i], fiy = fy[i], fiz = fz[i];

    // b128 loads of 4 neighbors per SoA plane (j0 % 4 == 0, planes 16B-aligned)
    const f4 xj = *(const f4*)(fx + j0);
    const f4 yj = *(const f4*)(fy + j0);
    const f4 zj = *(const f4*)(fz + j0);

    float sx[4], sy[4], sz[4], dd[4];
#pragma unroll
    for (int k = 0; k < 4; ++k) {
        // dfrac = frac_j - frac_i
        float dx = xj[k] - fix;
        float dy = yj[k] - fiy;
        float dz = zj[k] - fiz;
        // shift = -round(dfrac)  (RNE, matches jnp.round)
        float rx = rintf(dx), ry = rintf(dy), rz = rintf(dz);
        sx[k] = -rx; sy[k] = -ry; sz[k] = -rz;
        // u = dfrac + shift  in [-0.5, 0.5)
        float ux = dx - rx, uy = dy - ry, uz = dz - rz;
        // d = u @ cell (Cartesian)
        float cx = fmaf(ux, c00, fmaf(uy, c10, uz * c20));
        float cy = fmaf(ux, c01, fmaf(uy, c11, uz * c21));
        float cz = fmaf(ux, c02, fmaf(uy, c12, uz * c22));
        float d2 = fmaf(cx, cx, fmaf(cy, cy, cz * cz));
        bool mask = (d2 > 0.0f) && (d2 < rc2);
        dd[k] = mask ? sqrtf(d2) : 0.0f;
    }

    // dist row: one b128 NT store (i*n and j0 both multiples of 4)
    {
        f4 v; v.x = dd[0]; v.y = dd[1]; v.z = dd[2]; v.w = dd[3];
        f4* p = (f4*)(dist_out + (size_t)i * n + j0);
        __builtin_nontemporal_store(v, p);
    }
    // shift row: 12 floats -> three b128 NT stores, base 16B-aligned
    {
        f4* p = (f4*)(shift_out + ((size_t)i * n + j0) * 3);
        f4 a, b, c;
        a.x = sx[0]; a.y = sy[0]; a.z = sz[0]; a.w = sx[1];
        b.x = sy[1]; b.y = sz[1]; b.z = sx[2]; b.w = sy[2];
        c.x = sz[2]; c.y = sx[3]; c.z = sy[3]; c.w = sz[3];
        __builtin_nontemporal_store(a, p + 0);
        __builtin_nontemporal_store(b, p + 1);
        __builtin_nontemporal_store(c, p + 2);
    }
}

extern "C" void kernel_launch(void* const* d_in, const int* in_sizes, int n_in,
                              void* d_out, int out_size, void* d_ws, size_t ws_size,
                              hipStream_t stream) {
    const float* coord = (const float*)d_in[0];   // (n,3) f32
    const float* cell  = (const float*)d_in[1];   // (3,3) f32
    const int*   elems = (const int*)d_in[2];     // (n,)  int

    const int n = in_sizes[0] / 3;                // 6000
    float* out = (float*)d_out;
    float* dist_out  = out + (size_t)4 * n;
    float* shift_out = dist_out + (size_t)n * n;

    // Workspace layout: [0..15] hinv (padded), then SoA frac planes
    float* wsf  = (float*)d_ws;
    float* hinv = wsf;
    float* fx   = wsf + 16;
    float* fy   = fx + n;
    float* fz   = fy + n;

    inv3x3_kernel<<<1, 1, 0, stream>>>(cell, hinv);

    frac_wmma_kernel<<<(n + 15) / 16, 32, 0, stream>>>(coord, hinv, fx, fy, fz, n);

    onehot_kernel<<<(n + 255) / 256, 256, 0, stream>>>(elems, out, n);

    const int quads = (n + 3) / 4;                       // 1500
    dim3 grid((quads + 255) / 256, n);                   // (6, 6000)
    pair_kernel<<<grid, 256, 0, stream>>>(cell, fx, fy, fz,
                                          dist_out, shift_out, n, 25.0f);
}